// GIN_2731599200559
// MI455X (gfx1250) — compile-verified
//
#include <hip/hip_runtime.h>

typedef float v2f __attribute__((ext_vector_type(2)));
typedef float v8f __attribute__((ext_vector_type(8)));

#define NNODE 50000
#define NEDGE 625000
#define HDIM  128

// ---------------------------------------------------------------------------
// f32 WMMA GEMM: C[M,Nout] = act(A[M,K] @ W[K,Nout] + bias[Nout])
// Each wave32 computes a 64x16 output slab: 4 M-tiles of 16x16, sharing one
// B fragment per k-step across 4 independent V_WMMA_F32_16X16X4_F32 chains.
// Double-buffered: fragments for step k+4 are loaded before the 4 WMMAs of
// step k execute, keeping loads in flight behind the matrix ops.
//
// Fragment layouts (CDNA5 ISA 7.12.2):
//   A 16x4 : lanes 0-15 hold K=k0+0 (v0), k0+1 (v1); lanes 16-31 hold k0+2,3;
//            M = lane%16.
//   B 4x16 : lanes 0-15 hold K=k0+0 (v0), k0+1 (v1); lanes 16-31 hold k0+2,3;
//            N = lane%16.
//   C 16x16: v[r] is row M=r (lanes 0-15) / M=r+8 (lanes 16-31), N = lane%16.
//
// Ragged M handling: A row indices are CLAMPED to M-1 (an out-of-range A row
// only feeds the matching out-of-range D row, which is never stored), so the
// inner loop is completely divergence-free; stores are row-guarded.
// ---------------------------------------------------------------------------
__global__ void gemm_wmma_f32(const float* __restrict__ A,
                              const float* __restrict__ W,
                              const float* __restrict__ bias,
                              float* __restrict__ C,
                              int M, int K, int Nout, int relu)
{
    const int lane   = threadIdx.x & 31;
    const int waveId = threadIdx.x >> 5;
    const int tilesN = Nout >> 4;

    const int slab   = blockIdx.x * (blockDim.x >> 5) + waveId;
    const int slabM  = slab / tilesN;
    const int tileN  = slab - slabM * tilesN;
    const int m0     = slabM << 6;        // 64 rows per wave
    const int n0     = tileN << 4;
    if (m0 >= M) return;                  // wave-uniform exit

    const int half = lane >> 4;           // 0: K pair {0,1}, 1: K pair {2,3}
    const int l16  = lane & 15;
    const int nB   = n0 + l16;

    // Per-lane A row pointers for the 4 M-tiles (rows clamped into range).
    const float* aP[4];
    #pragma unroll
    for (int i = 0; i < 4; ++i) {
        int mA = m0 + (i << 4) + l16;
        mA = mA < M ? mA : M - 1;
        aP[i] = A + (size_t)mA * (size_t)K + (half << 1);
    }
    // Per-lane B pointer: row k0+2*half, column nB.
    const float* bP = W + (size_t)(half << 1) * (size_t)Nout + nB;
    const size_t bStep = (size_t)Nout * 4;

    v8f acc0 = {}, acc1 = {}, acc2 = {}, acc3 = {};

    // ---- prologue: load fragments for k0 = 0 ----
    v2f bc, a0c, a1c, a2c, a3c;
    bc.x = bP[0];
    bc.y = bP[Nout];
    a0c = *(const v2f*)(aP[0]);
    a1c = *(const v2f*)(aP[1]);
    a2c = *(const v2f*)(aP[2]);
    a3c = *(const v2f*)(aP[3]);

    // ---- main loop: prefetch k0+4 fragments, then 4 WMMAs on current ----
    #pragma unroll 4
    for (int k0 = 0; k0 < K - 4; k0 += 4) {
        const float* bN = bP + bStep * ((k0 >> 2) + 1);
        v2f bn, a0n, a1n, a2n, a3n;
        bn.x = bN[0];
        bn.y = bN[Nout];
        a0n = *(const v2f*)(aP[0] + k0 + 4);
        a1n = *(const v2f*)(aP[1] + k0 + 4);
        a2n = *(const v2f*)(aP[2] + k0 + 4);
        a3n = *(const v2f*)(aP[3] + k0 + 4);

        acc0 = __builtin_amdgcn_wmma_f32_16x16x4_f32(false, a0c, false, bc, (short)0, acc0, false, false);
        acc1 = __builtin_amdgcn_wmma_f32_16x16x4_f32(false, a1c, false, bc, (short)0, acc1, false, false);
        acc2 = __builtin_amdgcn_wmma_f32_16x16x4_f32(false, a2c, false, bc, (short)0, acc2, false, false);
        acc3 = __builtin_amdgcn_wmma_f32_16x16x4_f32(false, a3c, false, bc, (short)0, acc3, false, false);

        bc = bn; a0c = a0n; a1c = a1n; a2c = a2n; a3c = a3n;
    }

    // ---- epilogue: final k-step ----
    acc0 = __builtin_amdgcn_wmma_f32_16x16x4_f32(false, a0c, false, bc, (short)0, acc0, false, false);
    acc1 = __builtin_amdgcn_wmma_f32_16x16x4_f32(false, a1c, false, bc, (short)0, acc1, false, false);
    acc2 = __builtin_amdgcn_wmma_f32_16x16x4_f32(false, a2c, false, bc, (short)0, acc2, false, false);
    acc3 = __builtin_amdgcn_wmma_f32_16x16x4_f32(false, a3c, false, bc, (short)0, acc3, false, false);

    const float bv = bias[nB];
    v8f accs[4] = {acc0, acc1, acc2, acc3};
    #pragma unroll
    for (int i = 0; i < 4; ++i) {
        const int mBase = m0 + (i << 4) + (half << 3);
        #pragma unroll
        for (int r = 0; r < 8; ++r) {
            const int m = mBase + r;
            if (m < M) {
                float v = accs[i][r] + bv;
                if (relu) v = v > 0.0f ? v : 0.0f;
                C[(size_t)m * (size_t)Nout + (size_t)nB] = v;
            }
        }
    }
}

// ---------------------------------------------------------------------------
// Elementwise / scatter helpers
// ---------------------------------------------------------------------------
__global__ void fill_zero_kernel(float* __restrict__ p, long long n)
{
    long long i = (long long)blockIdx.x * blockDim.x + threadIdx.x;
    if (i < n) p[i] = 0.0f;
}

__global__ void degree_kernel(const int* __restrict__ dst,
                              float* __restrict__ deg, int E)
{
    int e = blockIdx.x * blockDim.x + threadIdx.x;
    if (e < E) unsafeAtomicAdd(&deg[dst[e]], 1.0f);
}

__global__ void invdeg_kernel(float* __restrict__ deg, int n)
{
    int i = blockIdx.x * blockDim.x + threadIdx.x;
    if (i < n) {
        float d = deg[i];
        deg[i] = 1.0f / fmaxf(d, 1.0f);
    }
}

// msg = x[src[e]] + ea[e]; agg[dst[e]] += msg   (4 channels per thread)
__global__ void msg_scatter_kernel(const float* __restrict__ x,
                                   const float* __restrict__ ea,
                                   const int* __restrict__ src,
                                   const int* __restrict__ dst,
                                   float* __restrict__ agg, int E)
{
    long long idx = (long long)blockIdx.x * blockDim.x + threadIdx.x;
    long long e = idx >> 5;                 // HDIM/4 = 32 chunks per edge
    int c = (int)(idx & 31) << 2;
    if (e < E) {
        const int s = src[e];
        const int d = dst[e];
        const float4 xv = *(const float4*)(x + (size_t)s * HDIM + c);
        const float4 ev = *(const float4*)(ea + (size_t)e * HDIM + c);
        float* ag = agg + (size_t)d * HDIM + c;
        unsafeAtomicAdd(ag + 0, xv.x + ev.x);
        unsafeAtomicAdd(ag + 1, xv.y + ev.y);
        unsafeAtomicAdd(ag + 2, xv.z + ev.z);
        unsafeAtomicAdd(ag + 3, xv.w + ev.w);
    }
}

// z = (1+eps[l])*x + agg*inv_deg[row]   (written in place into agg)
__global__ void gin_combine_kernel(const float* __restrict__ x,
                                   const float* __restrict__ invdeg,
                                   const float* __restrict__ eps, int l,
                                   float* __restrict__ agg, long long total)
{
    long long i = (long long)blockIdx.x * blockDim.x + threadIdx.x;
    if (i < total) {
        const int row = (int)(i >> 7);              // HDIM == 128
        const float scale = 1.0f + eps[l];
        agg[i] = scale * x[i] + agg[i] * invdeg[row];
    }
}

// ---------------------------------------------------------------------------
// Host-side orchestration
// ---------------------------------------------------------------------------
static inline int gemm_blocks(int M, int Nout)
{
    long long slabs = (long long)((M + 63) / 64) * (Nout / 16);
    return (int)((slabs + 7) / 8);   // 8 waves (slabs) per 256-thread block
}

extern "C" void kernel_launch(void* const* d_in, const int* in_sizes, int n_in,
                              void* d_out, int out_size, void* d_ws, size_t ws_size,
                              hipStream_t stream)
{
    const float* x_in      = (const float*)d_in[0];   // [N,128]
    const float* edge_attr = (const float*)d_in[1];   // [E,64]
    const int*   edge_idx  = (const int*)  d_in[2];   // [2,E]
    const float* node_W    = (const float*)d_in[3];   // [128,128]
    const float* node_b    = (const float*)d_in[4];   // [128]
    const float* edge_W    = (const float*)d_in[5];   // [64,128]
    const float* edge_b    = (const float*)d_in[6];   // [128]
    const float* W1        = (const float*)d_in[7];   // [L,128,256]
    const float* b1        = (const float*)d_in[8];   // [L,256]
    const float* W2        = (const float*)d_in[9];   // [L,256,128]
    const float* b2        = (const float*)d_in[10];  // [L,128]
    const float* eps       = (const float*)d_in[11];  // [L]
    const float* lin_W     = (const float*)d_in[12];  // [128,112]
    const float* lin_b     = (const float*)d_in[13];  // [112]
    float*       out       = (float*)d_out;           // [N,112]

    const int N = NNODE, E = NEDGE, H = HDIM;
    const int L = 3, OUTD = 112;

    const int* src = edge_idx;       // edge_index[0]
    const int* dst = edge_idx + E;   // edge_index[1]

    // ---- workspace carve (floats) ----
    float* ws = (float*)d_ws;
    size_t o = 0;
    float* xbuf   = ws + o; o += (size_t)N * H;       // current node features
    float* agg    = ws + o; o += (size_t)N * H;       // aggregation / z (in place)
    float* h1     = ws + o; o += (size_t)N * 2 * H;   // MLP hidden
    float* invdeg = ws + o; o += (size_t)N;           // degree -> inv degree
    float* ea     = ws + o; o += (size_t)E * H;       // encoded edge features
    (void)ws_size; (void)in_sizes; (void)n_in; (void)out_size;

    const int TB = 256;

    // 1) node encoder: xbuf = x_in @ node_W + node_b
    gemm_wmma_f32<<<gemm_blocks(N, H), TB, 0, stream>>>(
        x_in, node_W, node_b, xbuf, N, 128, H, 0);

    // 2) edge encoder: ea = edge_attr @ edge_W + edge_b
    gemm_wmma_f32<<<gemm_blocks(E, H), TB, 0, stream>>>(
        edge_attr, edge_W, edge_b, ea, E, 64, H, 0);

    // 3) in-degree -> inverse degree
    fill_zero_kernel<<<(N + TB - 1) / TB, TB, 0, stream>>>(invdeg, (long long)N);
    degree_kernel<<<(E + TB - 1) / TB, TB, 0, stream>>>(dst, invdeg, E);
    invdeg_kernel<<<(N + TB - 1) / TB, TB, 0, stream>>>(invdeg, N);

    const long long nh = (long long)N * H;
    const long long scat_threads = (long long)E * (H / 4);

    // 4) GIN layers
    for (int l = 0; l < L; ++l) {
        fill_zero_kernel<<<(int)((nh + TB - 1) / TB), TB, 0, stream>>>(agg, nh);
        msg_scatter_kernel<<<(int)((scat_threads + TB - 1) / TB), TB, 0, stream>>>(
            xbuf, ea, src, dst, agg, E);
        gin_combine_kernel<<<(int)((nh + TB - 1) / TB), TB, 0, stream>>>(
            xbuf, invdeg, eps, l, agg, nh);
        // h1 = relu(z @ W1[l] + b1[l])
        gemm_wmma_f32<<<gemm_blocks(N, 2 * H), TB, 0, stream>>>(
            agg, W1 + (size_t)l * H * 2 * H, b1 + (size_t)l * 2 * H,
            h1, N, H, 2 * H, 1);
        // x = relu(h1 @ W2[l] + b2[l])
        gemm_wmma_f32<<<gemm_blocks(N, H), TB, 0, stream>>>(
            h1, W2 + (size_t)l * 2 * H * H, b2 + (size_t)l * H,
            xbuf, N, 2 * H, H, 1);
    }

    // 5) output head: out = x @ lin_W + lin_b
    gemm_wmma_f32<<<gemm_blocks(N, OUTD), TB, 0, stream>>>(
        xbuf, lin_W, lin_b, out, N, H, OUTD, 0);
}